// GPTmm_14980845929099
// MI455X (gfx1250) — compile-verified
//
#include <hip/hip_runtime.h>
#include <hip/hip_bf16.h>

// Reference-faithful observation: logits depend ONLY on tok_table[idx] @ Wlm + blm
// (the transformer loop is dead code w.r.t. the returned value). We implement that
// single GEMM with V_WMMA_F32_16X16X4_F32 (fp32-exact, store-bandwidth bound).

#define D_EMB   1024
#define VOCAB   50257
#define ROWS    4096          // B*T = 4*1024

#define MT 2                  // 16-row M tiles per wave  -> 32 rows / wave
#define NT 4                  // 16-col N tiles per wave  -> 64 cols / wave
#define WAVES_PER_BLOCK 4     // block covers 128 rows x 64 cols

typedef __attribute__((ext_vector_type(2))) float v2f;
typedef __attribute__((ext_vector_type(8))) float v8f;

__global__ __launch_bounds__(32 * WAVES_PER_BLOCK)
void gpt_logits_wmma_f32(const int* __restrict__ idx,
                         const float* __restrict__ tok_table,
                         const float* __restrict__ Wlm,
                         const float* __restrict__ blm,
                         float* __restrict__ out)
{
    const int lane   = threadIdx.x & 31;
    const int wave   = threadIdx.x >> 5;
    const int lane16 = lane & 15;
    const int half   = lane >> 4;        // 0: lanes 0-15, 1: lanes 16-31

    const int m0 = blockIdx.x * (WAVES_PER_BLOCK * MT * 16) + wave * (MT * 16);
    const int n0 = blockIdx.y * (NT * 16);

    // ---- A-fragment row pointers (gather through idx), one per M tile ----
    // A 16x4 f32 layout: lane holds M = lane16, VGPR r = K = (k + r + 2*half)
    const float* aPtr[MT];
#pragma unroll
    for (int mi = 0; mi < MT; ++mi) {
        const int row    = m0 + mi * 16 + lane16;
        const int tokrow = idx[row];                       // 0 <= tokrow < VOCAB
        aPtr[mi] = tok_table + (size_t)tokrow * D_EMB + 2 * half;
    }

    // ---- B-fragment column pointers, one per N tile (clamped at vocab edge) ----
    // B 4x16 f32 layout: lane holds N = lane16, VGPR r = K row = (k + r + 2*half)
    const float* bPtr[NT];
    int ncol[NT];
#pragma unroll
    for (int t = 0; t < NT; ++t) {
        const int n = n0 + t * 16 + lane16;
        ncol[t]     = n;
        const int nc = (n < VOCAB) ? n : (VOCAB - 1);      // clamp: keep EXEC uniform
        bPtr[t] = Wlm + (size_t)(2 * half) * VOCAB + nc;
    }

    v8f acc[MT][NT];
#pragma unroll
    for (int mi = 0; mi < MT; ++mi)
#pragma unroll
        for (int t = 0; t < NT; ++t)
            acc[mi][t] = (v8f){0.f, 0.f, 0.f, 0.f, 0.f, 0.f, 0.f, 0.f};

    // ---- K loop: D_EMB / 4 = 256 steps, 8 WMMAs per step ----
    for (int k = 0; k < D_EMB; k += 4) {
        v2f a[MT];
#pragma unroll
        for (int mi = 0; mi < MT; ++mi)
            a[mi] = *(const v2f*)(aPtr[mi] + k);           // global_load_b64

        const size_t kOff = (size_t)k * VOCAB;
#pragma unroll
        for (int t = 0; t < NT; ++t) {
            const float* b = bPtr[t] + kOff;
            v2f bf;
            bf.x = b[0];                                   // K = k + 2*half
            bf.y = b[VOCAB];                               // K = k + 2*half + 1
#pragma unroll
            for (int mi = 0; mi < MT; ++mi) {
                acc[mi][t] = __builtin_amdgcn_wmma_f32_16x16x4_f32(
                    /*neg_a=*/false, a[mi],
                    /*neg_b=*/false, bf,
                    /*c_mod=*/(short)0, acc[mi][t],
                    /*reuse_a=*/false, /*reuse_b=*/false);
            }
        }
    }

    // ---- bias + store (C layout: VGPR r -> M = r + 8*half, N = lane16) ----
#pragma unroll
    for (int t = 0; t < NT; ++t) {
        const int n = ncol[t];
        if (n < VOCAB) {                                   // predicate only stores
            const float bias = blm[n];
#pragma unroll
            for (int mi = 0; mi < MT; ++mi) {
                const int mbase = m0 + mi * 16 + 8 * half;
#pragma unroll
                for (int r = 0; r < 8; ++r) {
                    out[(size_t)(mbase + r) * VOCAB + n] = acc[mi][t][r] + bias;
                }
            }
        }
    }
}

extern "C" void kernel_launch(void* const* d_in, const int* in_sizes, int n_in,
                              void* d_out, int out_size, void* d_ws, size_t ws_size,
                              hipStream_t stream) {
    (void)in_sizes; (void)n_in; (void)d_ws; (void)ws_size; (void)out_size;

    // setup_inputs order: 0=idx, 1=tok_table, 2=pos_table, ..., 18=Wlm, 19=blm
    const int*   idx       = (const int*)  d_in[0];
    const float* tok_table = (const float*)d_in[1];
    const float* Wlm       = (const float*)d_in[18];
    const float* blm       = (const float*)d_in[19];
    float*       out       = (float*)d_out;

    // M fast-varying so blocks sharing an Wlm N-strip (256 KB) run adjacently (L2 reuse).
    dim3 grid(ROWS / (WAVES_PER_BLOCK * MT * 16),        // 4096/128 = 32
              (VOCAB + NT * 16 - 1) / (NT * 16));        // ceil(50257/64) = 786
    dim3 block(32 * WAVES_PER_BLOCK);

    gpt_logits_wmma_f32<<<grid, block, 0, stream>>>(idx, tok_table, Wlm, blm, out);
}